// TimeAwareSelfAttention_78623671321129
// MI455X (gfx1250) — compile-verified
//
#include <hip/hip_runtime.h>
#include <hip/hip_bf16.h>

typedef float v2f __attribute__((ext_vector_type(2)));
typedef float v4f __attribute__((ext_vector_type(4)));
typedef float v8f __attribute__((ext_vector_type(8)));

#define Bsz 4
#define Lsz 256
#define Hsz 256
#define Dsz 64

// ---------------------------------------------------------------------------
// Kernel A: fused QKV projections, fp32 WMMA 16x16x4.
//   Qp = queries @ Wq^T + bq
//   Kp = keys    @ Wk^T + bk + abs_pos_K      (Q.K + Q.pK == Q.(K+pK))
//   Vp = values  @ Wv^T + bv + abs_pos_V      (attn@V + attn@pV == attn@(V+pV))
// One wave computes one 16x16 output tile; 1024 tiles per matrix, 3 matrices.
// ---------------------------------------------------------------------------
__global__ __launch_bounds__(256) void qkv_proj_kernel(
    const float* __restrict__ q_in, const float* __restrict__ k_in,
    const float* __restrict__ v_in,
    const float* __restrict__ Wq, const float* __restrict__ bq,
    const float* __restrict__ Wk, const float* __restrict__ bk,
    const float* __restrict__ Wv, const float* __restrict__ bv,
    const float* __restrict__ pK, const float* __restrict__ pV,
    float* __restrict__ Qp, float* __restrict__ Kp, float* __restrict__ Vp)
{
    const int lane = threadIdx.x & 31;
    const int widx = threadIdx.x >> 5;
    const int w    = blockIdx.x * 8 + widx;      // 0..3071
    const int mat  = w >> 10;                    // 0:Q 1:K 2:V
    const int tile = w & 1023;
    const int mt   = tile >> 4;                  // 0..63 (rows of B*L=1024)
    const int nt   = tile & 15;                  // 0..15 (cols of H=256)

    const float* X;  const float* W;  const float* bias;
    const float* pos; float* Y;
    if (mat == 0)      { X = q_in; W = Wq; bias = bq; pos = nullptr; Y = Qp; }
    else if (mat == 1) { X = k_in; W = Wk; bias = bk; pos = pK;      Y = Kp; }
    else               { X = v_in; W = Wv; bias = bv; pos = pV;      Y = Vp; }

    const int lr  = lane & 15;
    const int off = (lane >> 4) << 1;            // 0 for lanes 0-15, 2 for 16-31
    const int m   = mt * 16 + lr;                // A row (all lanes: M = lane%16)
    const int n   = nt * 16 + lr;                // B col

    v8f c = {};
    for (int k = 0; k < Hsz; k += 4) {
        // A 16x4 fp32: VGPR0 = K=(k+off), VGPR1 = K=(k+off+1)
        v2f a = *(const v2f*)(X + (size_t)m * Hsz + k + off);
        // B 4x16 fp32 = W^T chunk: B[kk][n] = W[n][kk]
        v2f bfr = *(const v2f*)(W + (size_t)n * Hsz + k + off);
        c = __builtin_amdgcn_wmma_f32_16x16x4_f32(
                false, a, false, bfr, (short)0, c, false, false);
    }

    // C/D layout: VGPR j -> row (j + 8*halfwave), col = lane%16
    const int rbase = mt * 16 + ((lane >> 4) << 3);
    const int col   = nt * 16 + lr;
    #pragma unroll
    for (int j = 0; j < 8; ++j) {
        const int row = rbase + j;
        float val = c[j] + bias[col];
        if (pos) val += pos[(size_t)row * Hsz + col];
        Y[(size_t)row * Hsz + col] = val;
    }
}

// ---------------------------------------------------------------------------
// Kernel B: one workgroup = (h, b, 16-query tile). 256 threads = 8 waves.
//  Phase 1: scores S[16][256] = Q@(K+pK)^T via WMMA (2 k-tiles per wave)
//  Phase 2: S += tK . Q   (nontemporal stream, 64KB of tK per query row)
//  Phase 3: mask (row time_mask | causal) + unnormalized softmax
//  Phase 4: out = exp(S)@(V+pV) via WMMA (4 n-tiles x 2 k-halves across waves)
//           + exp(S) . tV streamed by all 256 threads; divide by rowsum at end
// ---------------------------------------------------------------------------
__global__ __launch_bounds__(256) void attn_kernel(
    const float* __restrict__ Qp, const float* __restrict__ Kp,
    const float* __restrict__ Vp,
    const float* __restrict__ tK, const float* __restrict__ tV,
    const unsigned char* __restrict__ time_mask,
    float* __restrict__ out)
{
    const int qt = blockIdx.x;          // 0..15 : query tile
    const int b  = blockIdx.y;          // 0..3
    const int h  = blockIdx.z;          // 0..3
    const int q0 = qt * 16;
    const int tid  = threadIdx.x;
    const int lane = tid & 31;
    const int widx = tid >> 5;
    const int lr   = lane & 15;
    const int hlf  = lane >> 4;
    const int off  = hlf << 1;

    __shared__ float Qs[16][68];        // Q tile  (pad 4)
    __shared__ float S [16][260];       // scores / exp-scores (pad 4)
    __shared__ float Ouc[16][68];       // WMMA output accumulation
    __shared__ float red[16][16];
    __shared__ float rowsum[16], rowmax[16];

    // ---- load Q tile [16][64] from Qp[b, q0+r, h*64 + d] ----
    {
        const int r  = tid >> 4;
        const int d4 = (tid & 15) * 4;
        const v4f v = *(const v4f*)(Qp + ((size_t)(b * Lsz + q0 + r)) * Hsz + h * Dsz + d4);
        Qs[r][d4 + 0] = v.x; Qs[r][d4 + 1] = v.y;
        Qs[r][d4 + 2] = v.z; Qs[r][d4 + 3] = v.w;
    }
    __syncthreads();

    const float scale = 0.125f;         // 1/sqrt(64)

    // ---- Phase 1: S = scale * Q @ (K+pK)^T ;  wave w does k-tiles 2w, 2w+1 ----
    for (int kt = widx * 2; kt < widx * 2 + 2; ++kt) {
        const int kbase = kt * 16;
        const float* krow = Kp + ((size_t)(b * Lsz + kbase + lr)) * Hsz + h * Dsz;
        v8f c = {};
        for (int d0 = 0; d0 < Dsz; d0 += 4) {
            v2f a  = *(const v2f*)(&Qs[lr][d0 + off]);
            v2f bf = *(const v2f*)(krow + d0 + off);
            c = __builtin_amdgcn_wmma_f32_16x16x4_f32(
                    false, a, false, bf, (short)0, c, false, false);
        }
        #pragma unroll
        for (int j = 0; j < 8; ++j)
            S[j + hlf * 8][kbase + lr] = c[j] * scale;
    }
    __syncthreads();

    // ---- Phase 2: S[q][k] += scale * dot(tK[b,q0+q,k,h*64:+64], Q[q]) ----
    // Single-use stream -> nontemporal loads (th:NT), don't pollute L2.
    {
        const int k = tid;              // each thread owns one key column
        for (int q = 0; q < 16; ++q) {
            const v4f* tkp = (const v4f*)(tK +
                ((((size_t)b * Lsz + q0 + q) * Lsz + k) * Hsz) + h * Dsz);
            // prefetch this thread's chunk of the NEXT query row (+L*H floats)
            __builtin_prefetch(tkp + (size_t)(Lsz * Hsz / 4), 0, 0);
            float acc = 0.f;
            #pragma unroll
            for (int d4 = 0; d4 < 16; ++d4) {
                const v4f t = __builtin_nontemporal_load(tkp + d4);
                acc += t.x * Qs[q][d4 * 4 + 0] + t.y * Qs[q][d4 * 4 + 1]
                     + t.z * Qs[q][d4 * 4 + 2] + t.w * Qs[q][d4 * 4 + 3];
            }
            S[q][k] += acc * scale;
        }
    }
    __syncthreads();

    // ---- Phase 3: mask + unnormalized softmax (16 threads per row) ----
    {
        const int r = tid >> 4;         // row 0..15
        const int c = tid & 15;
        const bool rmask = time_mask[b * Lsz + q0 + r] != 0;
        const float NEGV = -4294967295.0f;        // -2^32 + 1
        float m = -1e38f;
        #pragma unroll
        for (int j = 0; j < 16; ++j) {
            const int k = j * 16 + c;             // stride-1 across threads
            float s = S[r][k];
            if (rmask || k > q0 + r) s = NEGV;    // future_time_mask == (k > q)
            S[r][k] = s;
            m = fmaxf(m, s);
        }
        red[r][c] = m;
        __syncthreads();
        if (c == 0) {
            float mm = red[r][0];
            #pragma unroll
            for (int j = 1; j < 16; ++j) mm = fmaxf(mm, red[r][j]);
            rowmax[r] = mm;
        }
        __syncthreads();
        const float rm = rowmax[r];
        float ps = 0.f;
        #pragma unroll
        for (int j = 0; j < 16; ++j) {
            const int k = j * 16 + c;
            const float e = __expf(S[r][k] - rm);
            S[r][k] = e;                          // store unnormalized weights
            ps += e;
        }
        red[r][c] = ps;
        __syncthreads();
        if (c == 0) {
            float ss = red[r][0];
            #pragma unroll
            for (int j = 1; j < 16; ++j) ss += red[r][j];
            rowsum[r] = ss;
        }
    }
    __syncthreads();

    // ---- Phase 4a: tV stream. thread owns d=tid%64, q in {qq, qq+4, qq+8, qq+12} ----
    // Fully coalesced (64 consecutive d per q-row), nontemporal.
    const int d  = tid & 63;
    const int qq = tid >> 6;
    float tvacc[4] = {0.f, 0.f, 0.f, 0.f};
    for (int k = 0; k < Lsz; ++k) {
        #pragma unroll
        for (int i = 0; i < 4; ++i) {
            const int q = qq + i * 4;
            const float tv = __builtin_nontemporal_load(
                tV + ((((size_t)b * Lsz + q0 + q) * Lsz + k) * Hsz) + h * Dsz + d);
            tvacc[i] += S[q][k] * tv;
        }
    }

    // ---- Phase 4b: WMMA  expS[16x256] @ (V+pV)[256x64] ----
    // wave w: n-tile = w%4 (16 cols of d), k-half = w/4 (128 of 256 keys)
    {
        const int ntile = widx & 3;
        const int khalf = widx >> 2;
        const int n0 = ntile * 16;
        v8f co = {};
        for (int kk = 0; kk < 128; kk += 4) {
            const int kb = khalf * 128 + kk;
            v2f a = *(const v2f*)(&S[lr][kb + off]);
            const float* vr = Vp + ((size_t)(b * Lsz + kb + off)) * Hsz + h * Dsz + n0 + lr;
            v2f bf;
            bf.x = vr[0];                 // row kb+off
            bf.y = vr[Hsz];               // row kb+off+1
            co = __builtin_amdgcn_wmma_f32_16x16x4_f32(
                    false, a, false, bf, (short)0, co, false, false);
        }
        if (khalf == 0) {
            #pragma unroll
            for (int j = 0; j < 8; ++j) Ouc[j + hlf * 8][n0 + lr] = co[j];
        }
        __syncthreads();
        if (khalf == 1) {
            #pragma unroll
            for (int j = 0; j < 8; ++j) Ouc[j + hlf * 8][n0 + lr] += co[j];
        }
    }
    __syncthreads();

    // ---- Final: (WMMA part + tV part) / rowsum -> out[b, q0+q, h*64+d] ----
    // Output written once, never re-read -> nontemporal store.
    #pragma unroll
    for (int i = 0; i < 4; ++i) {
        const int q = qq + i * 4;
        const float val = (Ouc[q][d] + tvacc[i]) / rowsum[q];
        __builtin_nontemporal_store(
            val, out + ((size_t)(b * Lsz + q0 + q)) * Hsz + h * Dsz + d);
    }
}

// ---------------------------------------------------------------------------
extern "C" void kernel_launch(void* const* d_in, const int* in_sizes, int n_in,
                              void* d_out, int out_size, void* d_ws, size_t ws_size,
                              hipStream_t stream) {
    const float* queries = (const float*)d_in[0];
    const float* keys    = (const float*)d_in[1];
    const float* values  = (const float*)d_in[2];
    const float* tK      = (const float*)d_in[3];
    const float* tV      = (const float*)d_in[4];
    const float* pK      = (const float*)d_in[5];
    const float* pV      = (const float*)d_in[6];
    const unsigned char* tmask = (const unsigned char*)d_in[7];
    // d_in[8] = future_time_mask: deterministic triu(k=1) -> computed as (k > q)
    const float* Wq = (const float*)d_in[9];
    const float* bq = (const float*)d_in[10];
    const float* Wk = (const float*)d_in[11];
    const float* bk = (const float*)d_in[12];
    const float* Wv = (const float*)d_in[13];
    const float* bv = (const float*)d_in[14];

    float* Qp = (float*)d_ws;                 // [B*L, H] = 1 MB
    float* Kp = Qp + (size_t)Bsz * Lsz * Hsz; // K + abs_pos_K
    float* Vp = Kp + (size_t)Bsz * Lsz * Hsz; // V + abs_pos_V
    float* out = (float*)d_out;

    // 3 matrices x 64x16 tiles = 3072 waves, 8 waves/block
    qkv_proj_kernel<<<384, 256, 0, stream>>>(
        queries, keys, values, Wq, bq, Wk, bk, Wv, bv, pK, pV, Qp, Kp, Vp);

    attn_kernel<<<dim3(16, Bsz, 4), 256, 0, stream>>>(
        Qp, Kp, Vp, tK, tV, tmask, out);
}